// DiscriminativeLoss_52639119179909
// MI455X (gfx1250) — compile-verified
//
#include <hip/hip_runtime.h>
#include <math.h>

// ---------------- constants (match reference) ----------------
#define DELTA_VAR  0.5f
#define DELTA_DIST 1.5f
#define VAR_W      1.0f
#define DIST_W     1.0f
#define REG_W      0.001f
#define EPS        1e-12f

typedef __attribute__((ext_vector_type(2))) float v2f;
typedef __attribute__((ext_vector_type(8))) float v8f;

// D = A(16x4) x B(4x16) + C, f32 everywhere.
// 8-arg pattern: (neg_a, A, neg_b, B, c_mod, C, reuse_a, reuse_b)
__device__ __forceinline__ v8f wmma_f32_16x16x4(v2f a, v2f b, v8f c) {
    return __builtin_amdgcn_wmma_f32_16x16x4_f32(
        false, a, false, b, (short)0, c, false, false);
}

// ---------------- kernel 0: zero workspace ----------------
__global__ void k_init(float* ws, int n) {
    int t = blockIdx.x * blockDim.x + threadIdx.x;
    if (t < n) ws[t] = 0.0f;
}

// ---------------- kernel 1: means_num[b][i][d] += g*p ; g_sum[b][i] ----------------
// Per-wave WMMA: A[m=i][k=pix] = g, B[k=pix][n=d] = p.
// K-slot -> pixel assignment is a free permutation as long as A and B agree, so
// each lane loads two float4 (pixels base+8*hi .. +7, 32B contiguous) and feeds
// 4 WMMAs per iteration (16 pixels):
//   wmma0: (q0.x,q0.y)  wmma1: (q0.z,q0.w)  wmma2: (q1.x,q1.y)  wmma3: (q1.z,q1.w)
__global__ void k_means_accum(const float* __restrict__ pred,
                              const float* __restrict__ tgt,
                              float* __restrict__ ws,
                              int L, int nbPerBatch, int pixPerBlock, int bs) {
    const int b    = blockIdx.x / nbPerBatch;
    const int blk  = blockIdx.x % nbPerBatch;
    const int tid  = threadIdx.x;
    const int wave = tid >> 5;
    const int lane = tid & 31;
    const int i16  = lane & 15;
    const int hi   = lane >> 4;

    __shared__ float s_means[256];
    __shared__ float s_gsum[16];
    s_means[tid] = 0.0f;
    if (tid < 16) s_gsum[tid] = 0.0f;
    __syncthreads();

    const size_t Ls = (size_t)L;
    const float* gplane = tgt  + ((size_t)b * 16 + i16) * Ls;
    const float* pplane = pred + ((size_t)b * 16 + i16) * Ls;

    const int pixPerWave = pixPerBlock >> 3;   // 8 waves / block, multiple of 16
    const int start = blk * pixPerBlock + wave * pixPerWave;
    const int end   = start + pixPerWave;

    v8f c = {};
    float gs = 0.0f;
    for (int base = start; base < end; base += 16) {
        const int off = base + 8 * hi;                       // 32B-aligned
        const float4 gq0 = *(const float4*)(gplane + off);
        const float4 gq1 = *(const float4*)(gplane + off + 4);
        const float4 pq0 = *(const float4*)(pplane + off);
        const float4 pq1 = *(const float4*)(pplane + off + 4);

        c = wmma_f32_16x16x4((v2f){gq0.x, gq0.y}, (v2f){pq0.x, pq0.y}, c);
        c = wmma_f32_16x16x4((v2f){gq0.z, gq0.w}, (v2f){pq0.z, pq0.w}, c);
        c = wmma_f32_16x16x4((v2f){gq1.x, gq1.y}, (v2f){pq1.x, pq1.y}, c);
        c = wmma_f32_16x16x4((v2f){gq1.z, gq1.w}, (v2f){pq1.z, pq1.w}, c);

        gs += (gq0.x + gq0.y + gq0.z + gq0.w) +
              (gq1.x + gq1.y + gq1.z + gq1.w);               // g_sum partial, instance i16
    }

    // C layout: vgpr r -> row (instance) = r + 8*hi, col (dim) = lane&15
#pragma unroll
    for (int r = 0; r < 8; ++r) {
        atomicAdd(&s_means[(r + 8 * hi) * 16 + i16], c[r]);
    }
    atomicAdd(&s_gsum[i16], gs);
    __syncthreads();

    float* means_num = ws;                   // [bs][16][16]
    float* g_sum     = ws + (size_t)bs * 256;
    atomicAdd(&means_num[b * 256 + tid], s_means[tid]);
    if (tid < 16) atomicAdd(&g_sum[b * 16 + tid], s_gsum[tid]);
}

// ---------------- kernel 2: finalize means ----------------
__global__ void k_finalize_means(float* ws, const int* __restrict__ n_objects, int bs) {
    int t = blockIdx.x * blockDim.x + threadIdx.x;
    if (t >= bs * 256) return;
    const int b = t >> 8;
    const int i = (t >> 4) & 15;
    float* means = ws;
    float* g_sum = ws + (size_t)bs * 256;
    float gs = g_sum[b * 16 + i];
    gs = gs < 1.0f ? 1.0f : gs;
    const float valid = (i < n_objects[b]) ? 1.0f : 0.0f;
    means[t] = means[t] / gs * valid;
}

// ---------------- kernel 3: variance term accumulation ----------------
// Per 16-pixel tile: pm = p(16x16) x means^T(16x16) via 4 chained WMMAs.
// A[m=pix][k=d]=p, B[k=d][n=i]=means[i][d] (loop invariant, preloaded).
__global__ void k_var_accum(const float* __restrict__ pred,
                            const float* __restrict__ tgt,
                            float* __restrict__ ws,
                            int L, int nbPerBatch, int pixPerBlock, int bs) {
    const int b    = blockIdx.x / nbPerBatch;
    const int blk  = blockIdx.x % nbPerBatch;
    const int tid  = threadIdx.x;
    const int wave = tid >> 5;
    const int lane = tid & 31;
    const int i16  = lane & 15;
    const int hi   = lane >> 4;

    __shared__ float s_var[16];
    if (tid < 16) s_var[tid] = 0.0f;
    __syncthreads();

    const float* means = ws;                                  // finalized
    const float* mrow  = means + (size_t)(b * 16 + i16) * 16; // means[b][i16][*]

    // Loop-invariant B operands + per-lane m2 (instance i16)
    v2f breg[4];
#pragma unroll
    for (int cc = 0; cc < 4; ++cc)
        breg[cc] = *(const v2f*)(mrow + 4 * cc + 2 * hi);
    float m2 = 0.0f;
#pragma unroll
    for (int d = 0; d < 16; ++d) { float v = mrow[d]; m2 += v * v; }

    const size_t Ls = (size_t)L;
    const float* pbase  = pred + (size_t)b * 16 * Ls;
    const float* gplane = tgt  + ((size_t)b * 16 + i16) * Ls;

    const int pixPerWave = pixPerBlock >> 3;
    const int start = blk * pixPerBlock + wave * pixPerWave;
    const int end   = start + pixPerWave;

    float accv = 0.0f;
    for (int base = start; base < end; base += 16) {
        v8f c = {};
        float p2h = 0.0f;
#pragma unroll
        for (int cc = 0; cc < 4; ++cc) {
            const int d0 = 4 * cc + 2 * hi;
            v2f a;
            a.x = pbase[(size_t)d0 * Ls + base + i16];        // coalesced over lanes 0-15 / 16-31
            a.y = pbase[(size_t)(d0 + 1) * Ls + base + i16];
            c = wmma_f32_16x16x4(a, breg[cc], c);
            p2h += a.x * a.x + a.y * a.y;
        }
        // p2 for pixel (base + lane&15): combine the two d-halves across half-waves
        float p2full = p2h + __shfl_xor(p2h, 16, 32);

        // g for this lane's C rows: plane i16, pixels base+8*hi .. +7 (32B-aligned)
        const float4 g0 = *(const float4*)(gplane + base + 8 * hi);
        const float4 g1 = *(const float4*)(gplane + base + 8 * hi + 4);
        const float gvals[8] = {g0.x, g0.y, g0.z, g0.w, g1.x, g1.y, g1.z, g1.w};

#pragma unroll
        for (int r = 0; r < 8; ++r) {
            // C vgpr r -> pixel = base + r + 8*hi, instance = lane&15
            const float p2r = __shfl(p2full, r + 8 * hi, 32); // broadcast p2 of that pixel
            float sq = p2r - 2.0f * c[r] + m2;
            sq = sq < EPS ? EPS : sq;
            float h = sqrtf(sq) - DELTA_VAR;
            h = h < 0.0f ? 0.0f : h;
            accv += h * h * gvals[r];
        }
    }

    accv += __shfl_xor(accv, 16, 32);     // combine pixel halves (same instance)
    if (lane < 16) atomicAdd(&s_var[i16], accv);
    __syncthreads();

    float* var_acc = ws + (size_t)bs * 256 + (size_t)bs * 16;
    if (tid < 16) atomicAdd(&var_acc[b * 16 + tid], s_var[tid]);
}

// ---------------- kernel 4: distance + reg terms, final scalar ----------------
// One block; wave w handles batch w.
__global__ void k_final(const float* __restrict__ ws,
                        const int* __restrict__ n_objects,
                        float* __restrict__ out, int bs) {
    const int tid  = threadIdx.x;
    const int b    = tid >> 5;
    const int lane = tid & 31;

    const float* means   = ws;
    const float* g_sum   = ws + (size_t)bs * 256;
    const float* var_acc = ws + (size_t)bs * 256 + (size_t)bs * 16;

    __shared__ float s_part[32];

    const int   no  = n_objects[b];
    const float nof = (float)no;

    // var + reg partials: lanes 0-15 own instance i = lane
    float var_p = 0.0f, reg_p = 0.0f;
    if (lane < 16) {
        const int i = lane;
        const float valid = (i < no) ? 1.0f : 0.0f;
        float gs = g_sum[b * 16 + i];
        gs = gs < 1.0f ? 1.0f : gs;
        var_p = var_acc[b * 16 + i] / gs * valid;
        float m2 = 0.0f;
        const float* mr = means + (size_t)(b * 16 + i) * 16;
#pragma unroll
        for (int d = 0; d < 16; ++d) m2 += mr[d] * mr[d];
        reg_p = sqrtf(m2 < EPS ? EPS : m2) * valid;
    }

    // distance: 256 ordered (i,j) pairs, 8 per lane
    float dist_p = 0.0f;
    for (int t = 0; t < 8; ++t) {
        const int pidx = lane + 32 * t;
        const int i = pidx >> 4, j = pidx & 15;
        if (i != j && i < no && j < no) {
            const float* mi = means + (size_t)(b * 16 + i) * 16;
            const float* mj = means + (size_t)(b * 16 + j) * 16;
            float psq = 0.0f;
#pragma unroll
            for (int d = 0; d < 16; ++d) { float df = mi[d] - mj[d]; psq += df * df; }
            psq = psq < EPS ? EPS : psq;
            float h = 2.0f * DELTA_DIST - sqrtf(psq);
            h = h < 0.0f ? 0.0f : h;
            dist_p += h * h;
        }
    }

    // wave-wide reductions
    for (int s = 16; s > 0; s >>= 1) {
        var_p  += __shfl_xor(var_p,  s, 32);
        reg_p  += __shfl_xor(reg_p,  s, 32);
        dist_p += __shfl_xor(dist_p, s, 32);
    }

    if (lane == 0) {
        float denom = nof * (nof - 1.0f);
        denom = denom < 1.0f ? 1.0f : denom;
        const float multi = (no > 1) ? 1.0f : 0.0f;
        const float var_b  = var_p / nof;
        const float dist_b = dist_p / denom * multi;
        const float reg_b  = reg_p / nof;
        s_part[b] = VAR_W * var_b + DIST_W * dist_b + REG_W * reg_b;
    }
    __syncthreads();
    if (tid == 0) {
        float s = 0.0f;
        for (int k = 0; k < bs; ++k) s += s_part[k];
        out[0] = s / (float)bs;
    }
}

// ---------------- host launch ----------------
extern "C" void kernel_launch(void* const* d_in, const int* in_sizes, int n_in,
                              void* d_out, int out_size, void* d_ws, size_t ws_size,
                              hipStream_t stream) {
    const float* pred = (const float*)d_in[0];   // (bs,16,H,W) f32
    const float* tgt  = (const float*)d_in[1];   // (bs,16,H,W) f32 one-hot
    const int*   nobj = (const int*)d_in[2];     // (bs,) i32
    float* ws  = (float*)d_ws;
    float* out = (float*)d_out;

    const int bs = in_sizes[2];                  // 8
    const int L  = in_sizes[0] / (bs * 16);      // 512*512 = 262144

    // workspace: means[bs*256] | g_sum[bs*16] | var_acc[bs*16]
    const int wsN = bs * 256 + bs * 16 + bs * 16;
    k_init<<<(wsN + 255) / 256, 256, 0, stream>>>(ws, wsN);

    const int pixPerBlock = 2048;                // 256 px per wave, 8 waves/block
    const int nb = L / pixPerBlock;              // 128 blocks per batch

    k_means_accum<<<bs * nb, 256, 0, stream>>>(pred, tgt, ws, L, nb, pixPerBlock, bs);
    k_finalize_means<<<(bs * 256 + 255) / 256, 256, 0, stream>>>(ws, nobj, bs);
    k_var_accum<<<bs * nb, 256, 0, stream>>>(pred, tgt, ws, L, nb, pixPerBlock, bs);
    k_final<<<1, bs * 32, 0, stream>>>(ws, nobj, out, bs);
}